// SpatialGNN_68496138437285
// MI455X (gfx1250) — compile-verified
//
#include <hip/hip_runtime.h>
#include <hip/hip_bf16.h>

// ---------------------------------------------------------------------------
// Graph Transformer conv (PyG TransformerConv style), 3 layers, fp32.
// N=50000 nodes, E=800000 edges, HID=64, HEADS=4, D=16.
// Dense projections use V_WMMA_F32_16X16X4_F32 (full fp32 precision, wave32).
// Edge phase is memory/atomic bound: gathers + scatter softmax + scatter sum.
// ---------------------------------------------------------------------------

#define GNN_N 50000
#define GNN_E 800000
#define GNN_HID 64
#define GNN_HEADS 4
#define GNN_D 16

typedef float v2f __attribute__((ext_vector_type(2)));
typedef float v8f __attribute__((ext_vector_type(8)));

// ---------------------------------------------------------------------------
// WMMA GEMM: out[N,64] = A[N,64] @ W[64,64] + bias[64]
// One wave per 16-row block. 16 K-steps of V_WMMA_F32_16X16X4_F32 per N-tile,
// 4 N-tiles of 16 columns. N = 50000 = 3125 * 16 exactly, so every
// participating wave has EXEC all-1s (whole waves retire on the bounds check).
// ---------------------------------------------------------------------------
__global__ __launch_bounds__(256) void gnn_gemm64_wmma(
    const float* __restrict__ A, const float* __restrict__ W,
    const float* __restrict__ bias, float* __restrict__ out, int nBlocks) {
  const int wave = blockIdx.x * (blockDim.x >> 5) + (threadIdx.x >> 5);
  if (wave >= nBlocks) return;  // whole-wave uniform exit
  const int lane = threadIdx.x & 31;
  const int lm = lane & 15;   // row (A) / col (B) within the 16-wide tile
  const int lh = lane >> 4;   // half-select: lanes 16-31 carry K+2 / M+8
  const int row0 = wave * 16;

  v8f acc[4];
  acc[0] = v8f{}; acc[1] = v8f{}; acc[2] = v8f{}; acc[3] = v8f{};

  // A fragment source: row (row0+lm), columns 4*kk + 2*lh + {0,1}
  const float* arow = A + (row0 + lm) * GNN_HID + 2 * lh;

#pragma unroll
  for (int kk = 0; kk < 16; ++kk) {
    const float2 at = *(const float2*)(arow + 4 * kk);
    v2f a; a.x = at.x; a.y = at.y;
    // B fragment: rows 4*kk + 2*lh + {0,1}, column (n0 + lm)
    const float* wrow = W + (4 * kk + 2 * lh) * GNN_HID + lm;
#pragma unroll
    for (int nt = 0; nt < 4; ++nt) {
      v2f b;
      b.x = wrow[nt * 16];            // K = 4kk + 2lh
      b.y = wrow[GNN_HID + nt * 16];  // K = 4kk + 2lh + 1
      acc[nt] = __builtin_amdgcn_wmma_f32_16x16x4_f32(
          /*neg_a=*/false, a, /*neg_b=*/false, b,
          /*c_mod=*/(short)0, acc[nt], /*reuse_a=*/false, /*reuse_b=*/false);
    }
  }

  // C/D layout: VGPR r holds M = r + 8*lh, N = lm (per CDNA5 ISA)
#pragma unroll
  for (int nt = 0; nt < 4; ++nt) {
    const float bv = bias[nt * 16 + lm];
#pragma unroll
    for (int r = 0; r < 8; ++r) {
      out[(row0 + 8 * lh + r) * GNN_HID + nt * 16 + lm] = acc[nt][r] + bv;
    }
  }
}

// ---------------------------------------------------------------------------
// Float atomic max via int ordering trick (works for mixed signs, -inf init).
// ---------------------------------------------------------------------------
__device__ __forceinline__ void atomicMaxFloat(float* addr, float value) {
  if (value >= 0.0f)
    atomicMax((int*)addr, __float_as_int(value));
  else
    atomicMin((unsigned int*)addr, __float_as_uint(value));
}

// m = -inf, denom = 0 for the scatter softmax
__global__ void gnn_init_softmax(float* __restrict__ m, float* __restrict__ denom) {
  const int t = blockIdx.x * blockDim.x + threadIdx.x;
  if (t >= GNN_N * GNN_HEADS) return;
  m[t] = -INFINITY;
  denom[t] = 0.0f;
}

// ---------------------------------------------------------------------------
// alpha[e,h] = dot(q[dst], k[src] + ea*We + be) / sqrt(D); atomic max per dst
// One thread per (edge, head); 16-dim dot via float4 loads (rows 64B aligned).
// ---------------------------------------------------------------------------
__global__ void gnn_edge_alpha(
    const float* __restrict__ q, const float* __restrict__ k,
    const int* __restrict__ src, const int* __restrict__ dst,
    const float* __restrict__ ea, const float* __restrict__ We,
    const float* __restrict__ be, float* __restrict__ alpha,
    float* __restrict__ m) {
  const int t = blockIdx.x * blockDim.x + threadIdx.x;
  if (t >= GNN_E * GNN_HEADS) return;
  const int e = t >> 2;
  const int h = t & 3;
  const int s = src[e];
  const int d = dst[e];
  const float a = ea[e];

  const float4* qp = (const float4*)(q + d * GNN_HID + h * GNN_D);
  const float4* kp = (const float4*)(k + s * GNN_HID + h * GNN_D);
  const float4* wp = (const float4*)(We + h * GNN_D);
  const float4* bp = (const float4*)(be + h * GNN_D);

  float dot = 0.0f;
#pragma unroll
  for (int i = 0; i < 4; ++i) {
    const float4 qv = qp[i];
    const float4 kv = kp[i];
    const float4 wv = wp[i];
    const float4 bv = bp[i];
    dot += qv.x * (kv.x + a * wv.x + bv.x);
    dot += qv.y * (kv.y + a * wv.y + bv.y);
    dot += qv.z * (kv.z + a * wv.z + bv.z);
    dot += qv.w * (kv.w + a * wv.w + bv.w);
  }
  const float al = dot * 0.25f;  // 1/sqrt(16)
  alpha[t] = al;
  atomicMaxFloat(&m[d * GNN_HEADS + h], al);
}

// ex = exp(alpha - m[dst]); denom[dst] += ex (atomic)
__global__ void gnn_edge_exp(
    const int* __restrict__ dst, const float* __restrict__ m,
    float* __restrict__ alpha, float* __restrict__ denom) {
  const int t = blockIdx.x * blockDim.x + threadIdx.x;
  if (t >= GNN_E * GNN_HEADS) return;
  const int e = t >> 2;
  const int h = t & 3;
  const int d = dst[e];
  const float ex = __expf(alpha[t] - m[d * GNN_HEADS + h]);
  alpha[t] = ex;
  atomicAdd(&denom[d * GNN_HEADS + h], ex);
}

// ---------------------------------------------------------------------------
// agg[dst] += (v[src] + e) * ex/(denom+1e-16). One thread per (edge, channel).
// agg was pre-initialized to the root skip (h @ Ws + bs) by the WMMA GEMM.
// ---------------------------------------------------------------------------
__global__ void gnn_edge_scatter(
    const float* __restrict__ v, const int* __restrict__ src,
    const int* __restrict__ dst, const float* __restrict__ ea,
    const float* __restrict__ We, const float* __restrict__ be,
    const float* __restrict__ alpha, const float* __restrict__ denom,
    float* __restrict__ agg) {
  const int t = blockIdx.x * blockDim.x + threadIdx.x;
  if (t >= GNN_E * GNN_HID) return;
  const int e = t >> 6;
  const int c = t & 63;
  const int h = c >> 4;
  const int s = src[e];
  const int d = dst[e];
  __builtin_prefetch(&v[s * GNN_HID + c], 0, 1);  // global_prefetch_b8
  const float w = alpha[e * GNN_HEADS + h] /
                  (denom[d * GNN_HEADS + h] + 1e-16f);
  const float val = (v[s * GNN_HID + c] + ea[e] * We[c] + be[c]) * w;
  atomicAdd(&agg[d * GNN_HID + c], val);
}

// ---------------------------------------------------------------------------
// out = LayerNorm(h + agg) * g + b. One wave32 per row (2 channels / lane),
// reductions with __shfl_xor across the 32-lane wave.
// ---------------------------------------------------------------------------
__global__ __launch_bounds__(256) void gnn_layernorm(
    const float* __restrict__ h, const float* __restrict__ agg,
    const float* __restrict__ g, const float* __restrict__ b,
    float* __restrict__ out) {
  const int row = blockIdx.x * (blockDim.x >> 5) + (threadIdx.x >> 5);
  if (row >= GNN_N) return;
  const int lane = threadIdx.x & 31;

  const float2 hv = ((const float2*)(h + row * GNN_HID))[lane];
  const float2 av = ((const float2*)(agg + row * GNN_HID))[lane];
  const float x0 = hv.x + av.x;
  const float x1 = hv.y + av.y;

  float sum = x0 + x1;
  float sq = x0 * x0 + x1 * x1;
#pragma unroll
  for (int off = 16; off > 0; off >>= 1) {
    sum += __shfl_xor(sum, off, 32);
    sq += __shfl_xor(sq, off, 32);
  }
  const float mu = sum * (1.0f / GNN_HID);
  const float var = sq * (1.0f / GNN_HID) - mu * mu;
  const float inv = rsqrtf(var + 1e-5f);

  const float2 gv = ((const float2*)g)[lane];
  const float2 bv = ((const float2*)b)[lane];
  float2 o;
  o.x = (x0 - mu) * inv * gv.x + bv.x;
  o.y = (x1 - mu) * inv * gv.y + bv.y;
  ((float2*)(out + row * GNN_HID))[lane] = o;
}

// ---------------------------------------------------------------------------
// Host-side orchestration
// ---------------------------------------------------------------------------
extern "C" void kernel_launch(void* const* d_in, const int* in_sizes, int n_in,
                              void* d_out, int out_size, void* d_ws, size_t ws_size,
                              hipStream_t stream) {
  (void)in_sizes; (void)n_in; (void)out_size; (void)ws_size;

  const float* x    = (const float*)d_in[0];
  const int*   eidx = (const int*)d_in[1];     // (2,E): row0=src, row1=dst
  const float* ea   = (const float*)d_in[2];   // (E,1)
  const float* in_W = (const float*)d_in[3];
  const float* in_b = (const float*)d_in[4];
  const float* Wq   = (const float*)d_in[5];   // (L,64,64)
  const float* bq   = (const float*)d_in[6];   // (L,64)
  const float* Wk   = (const float*)d_in[7];
  const float* bk   = (const float*)d_in[8];
  const float* Wv   = (const float*)d_in[9];
  const float* bv   = (const float*)d_in[10];
  const float* We   = (const float*)d_in[11];  // (L,1,64)
  const float* be   = (const float*)d_in[12];
  const float* Ws   = (const float*)d_in[13];
  const float* bs   = (const float*)d_in[14];
  const float* lng  = (const float*)d_in[15];
  const float* lnb  = (const float*)d_in[16];

  const int* src = eidx;
  const int* dst = eidx + GNN_E;

  // Workspace layout (floats)
  float* ws = (float*)d_ws;
  const size_t NF = (size_t)GNN_N * GNN_HID;
  float* h     = ws;
  float* q     = ws + 1 * NF;
  float* k     = ws + 2 * NF;
  float* v     = ws + 3 * NF;
  float* agg   = ws + 4 * NF;
  float* alpha = ws + 5 * NF;                        // E*4 floats
  float* m     = alpha + (size_t)GNN_E * GNN_HEADS;  // N*4
  float* denom = m + (size_t)GNN_N * GNN_HEADS;      // N*4

  const int nRowBlocks = GNN_N / 16;  // 3125
  const dim3 gemmGrid((nRowBlocks + 7) / 8);
  const dim3 blk256(256);
  const int ehThreads = GNN_E * GNN_HEADS;
  const dim3 ehGrid((ehThreads + 255) / 256);
  const int ecThreads = GNN_E * GNN_HID;
  const dim3 ecGrid((ecThreads + 255) / 256);
  const dim3 initGrid((GNN_N * GNN_HEADS + 255) / 256);
  const dim3 lnGrid((GNN_N + 7) / 8);

  // Input projection: h = x @ in_W + in_b
  gnn_gemm64_wmma<<<gemmGrid, blk256, 0, stream>>>(x, in_W, in_b, h, nRowBlocks);

  for (int l = 0; l < 3; ++l) {
    const float* Wq_l = Wq + (size_t)l * GNN_HID * GNN_HID;
    const float* Wk_l = Wk + (size_t)l * GNN_HID * GNN_HID;
    const float* Wv_l = Wv + (size_t)l * GNN_HID * GNN_HID;
    const float* Ws_l = Ws + (size_t)l * GNN_HID * GNN_HID;
    const float* bq_l = bq + (size_t)l * GNN_HID;
    const float* bk_l = bk + (size_t)l * GNN_HID;
    const float* bv_l = bv + (size_t)l * GNN_HID;
    const float* bs_l = bs + (size_t)l * GNN_HID;
    const float* We_l = We + (size_t)l * GNN_HID;
    const float* be_l = be + (size_t)l * GNN_HID;
    const float* g_l  = lng + (size_t)l * GNN_HID;
    const float* b_l  = lnb + (size_t)l * GNN_HID;

    gnn_gemm64_wmma<<<gemmGrid, blk256, 0, stream>>>(h, Wq_l, bq_l, q, nRowBlocks);
    gnn_gemm64_wmma<<<gemmGrid, blk256, 0, stream>>>(h, Wk_l, bk_l, k, nRowBlocks);
    gnn_gemm64_wmma<<<gemmGrid, blk256, 0, stream>>>(h, Wv_l, bv_l, v, nRowBlocks);
    // Root skip written straight into the aggregation buffer
    gnn_gemm64_wmma<<<gemmGrid, blk256, 0, stream>>>(h, Ws_l, bs_l, agg, nRowBlocks);

    gnn_init_softmax<<<initGrid, blk256, 0, stream>>>(m, denom);
    gnn_edge_alpha<<<ehGrid, blk256, 0, stream>>>(q, k, src, dst, ea, We_l, be_l,
                                                  alpha, m);
    gnn_edge_exp<<<ehGrid, blk256, 0, stream>>>(dst, m, alpha, denom);
    gnn_edge_scatter<<<ecGrid, blk256, 0, stream>>>(v, src, dst, ea, We_l, be_l,
                                                    alpha, denom, agg);

    float* outp = (l == 2) ? (float*)d_out : h;
    gnn_layernorm<<<lnGrid, blk256, 0, stream>>>(h, agg, g_l, b_l, outp);
  }
}